// YOLACTLoss_17557826306309
// MI455X (gfx1250) — compile-verified
//
#include <hip/hip_runtime.h>
#include <math.h>

// ---------------- problem constants ----------------
#define BB 8
#define AA 19248
#define GG 32
#define CC 81
#define PP 32
#define PH_ 138
#define PW_ 138
#define HW (PH_*PW_)          // 19044
#define KPOS 128
#define NBLK ((AA + 255)/256)     // 76  (match kernel blocks)
#define NB_CB (AA/8)              // 2406 (clsbox: 8 anchors per 256-thr block)
#define NCT ((HW + 15)/16)        // 1191 column tiles of 16

#define INPUT_SIZE 550.0f
#define POS_THR 0.5f
#define NEG_THR 0.4f
#define ALPHA_ 0.25f
#define NEG_POS_RATIO 3
#define CLS_W 1.0f
#define BOX_W 1.5f
#define MASK_W 6.125f
#define VAR_XY 0.1f
#define VAR_WH 0.2f
#define EPSF 1e-6f

typedef __attribute__((ext_vector_type(2))) float v2f;
typedef __attribute__((ext_vector_type(8))) float v8f;

// ---------------- scratch layout (bytes) ----------------
#define OFF_PACKED   0                                   // u64 [B*G]
#define OFF_BIOU     (OFF_PACKED + BB*GG*8)              // f32 [B*A]
#define OFF_BIDX     (OFF_BIOU   + BB*AA*4)              // i32 [B*A]
#define OFF_POS      (OFF_BIDX   + BB*AA*4)              // i32 [B*A]
#define OFF_NEG      (OFF_POS    + BB*AA*4)              // i32 [B*A]
#define OFF_NUMPOS   (OFF_NEG    + BB*AA*4)              // i32 [B]
#define OFF_CLSPART  (OFF_NUMPOS + BB*4)                 // f32 [B*NB_CB]
#define OFF_BOXPART  (OFF_CLSPART+ BB*NB_CB*4)           // f32 [B*NB_CB]
#define OFF_SELIDX   (OFF_BOXPART+ BB*NB_CB*4)           // i32 [B*K]
#define OFF_SELW     (OFF_SELIDX + BB*KPOS*4)            // f32 [B*K]
#define OFF_SELMI    (OFF_SELW   + BB*KPOS*4)            // i32 [B*K]
#define OFF_SELCOEF  (OFF_SELMI  + BB*KPOS*4)            // f32 [B*K*P]
#define OFF_ROWSUM   (OFF_SELCOEF+ BB*KPOS*PP*4)         // f32 [B*K]
#define WS_USED      (OFF_ROWSUM + BB*KPOS*4)

// ================= kernel 1: anchor/GT matching =================
__global__ void match_kernel(const float* __restrict__ anchors,
                             const float* __restrict__ gt_boxes,
                             float* __restrict__ best_iou,
                             int* __restrict__ best_idx,
                             int* __restrict__ pos,
                             unsigned long long* __restrict__ packed) {
    int b = blockIdx.y;
    int a = blockIdx.x * 256 + threadIdx.x;
    __shared__ float4 gtb[GG];
    if (threadIdx.x < GG) {
        const float4 g4 = ((const float4*)gt_boxes)[(size_t)b*GG + threadIdx.x];
        const float inv = 1.0f / INPUT_SIZE;
        gtb[threadIdx.x] = make_float4(g4.x*inv, g4.y*inv, g4.z*inv, g4.w*inv);
    }
    __syncthreads();
    if (a >= AA) return;
    float4 an = ((const float4*)anchors)[a];
    float ax1 = an.x - 0.5f*an.z, ay1 = an.y - 0.5f*an.w;
    float ax2 = an.x + 0.5f*an.z, ay2 = an.y + 0.5f*an.w;
    float areaA = (ax2-ax1)*(ay2-ay1);
    float bestv = -1.0f; int bestg = 0;
    for (int g = 0; g < GG; ++g) {
        float4 gb = gtb[g];
        float ix = fminf(ax2,gb.z) - fmaxf(ax1,gb.x); ix = fmaxf(ix, 0.0f);
        float iy = fminf(ay2,gb.w) - fmaxf(ay1,gb.y); iy = fmaxf(iy, 0.0f);
        float inter = ix*iy;
        float areaB = (gb.z-gb.x)*(gb.w-gb.y);
        float uni = fmaxf(areaA + areaB - inter, EPSF);
        float iou = inter / uni;
        if (iou > bestv) { bestv = iou; bestg = g; }
        unsigned long long key =
            ((unsigned long long)__float_as_uint(iou) << 32) |
            (unsigned long long)(0x7FFFFFFFu - (unsigned)a);
        atomicMax(&packed[b*GG + g], key);
    }
    best_iou[(size_t)b*AA + a] = bestv;
    best_idx[(size_t)b*AA + a] = bestg;
    pos[(size_t)b*AA + a] = (bestv >= POS_THR) ? 1 : 0;
}

// ================= kernel 2: per-GT scatter (sequential, last-wins) ===========
__global__ void scatter_kernel(const unsigned long long* __restrict__ packed,
                               int* __restrict__ pos, int* __restrict__ best_idx) {
    int b = blockIdx.x;
    if (threadIdx.x != 0) return;
    for (int g = 0; g < GG; ++g) {
        unsigned long long key = packed[b*GG + g];
        int a = (int)(0x7FFFFFFFu - (unsigned)(key & 0xFFFFFFFFull));
        pos[(size_t)b*AA + a] = 1;
        best_idx[(size_t)b*AA + a] = g;
    }
}

// ================= kernel 3: num_pos + hard-negative mining (block scan) ======
__global__ void scan_kernel(const float* __restrict__ best_iou,
                            const int* __restrict__ pos,
                            int* __restrict__ neg, int* __restrict__ numpos) {
    int b = blockIdx.x; int tid = threadIdx.x;
    __shared__ int s[256];
    __shared__ int sh_npos;
    __shared__ int sh_carry;
    int cnt = 0;
    for (int a = tid; a < AA; a += 256) cnt += pos[(size_t)b*AA + a];
    s[tid] = cnt; __syncthreads();
    for (int off = 128; off > 0; off >>= 1) {
        if (tid < off) s[tid] += s[tid + off];
        __syncthreads();
    }
    if (tid == 0) { sh_npos = s[0]; numpos[b] = s[0]; sh_carry = 0; }
    __syncthreads();
    int limit = sh_npos * NEG_POS_RATIO;
    for (int chunk = 0; chunk < NBLK; ++chunk) {
        int a = chunk*256 + tid;
        int v = 0;
        if (a < AA)
            v = (best_iou[(size_t)b*AA + a] < NEG_THR && !pos[(size_t)b*AA + a]) ? 1 : 0;
        s[tid] = v; __syncthreads();
        for (int off = 1; off < 256; off <<= 1) {
            int t = (tid >= off) ? s[tid - off] : 0;
            __syncthreads();
            s[tid] += t;
            __syncthreads();
        }
        int myinc = s[tid];
        int total = s[255];
        int c = sh_carry;
        if (a < AA) neg[(size_t)b*AA + a] = (v && (c + myinc <= limit)) ? 1 : 0;
        __syncthreads();
        if (tid == 0) sh_carry = c + total;
        __syncthreads();
    }
}

// ====== kernel 4: focal cls + smooth-L1 box loss, wave-per-anchor (wave32) ====
__global__ void clsbox_kernel(const float* __restrict__ class_preds,
                              const float* __restrict__ box_preds,
                              const float* __restrict__ anchors,
                              const float* __restrict__ gt_boxes,
                              const int*   __restrict__ gt_labels,
                              const int*   __restrict__ best_idx,
                              const int*   __restrict__ pos,
                              const int*   __restrict__ neg,
                              float* __restrict__ cls_part,
                              float* __restrict__ box_part) {
    int b  = blockIdx.y;
    int wv = threadIdx.x >> 5;
    int ln = threadIdx.x & 31;
    int a  = blockIdx.x * 8 + wv;          // AA divisible by 8: no tail
    __shared__ float w1[8], w2[8];

    const float* x = class_preds + ((size_t)b*AA + a) * CC;
    // lane-cooperative max over 81 logits (coalesced)
    float mx = -3.4e38f;
    for (int c = ln; c < CC; c += 32) mx = fmaxf(mx, x[c]);
    for (int off = 16; off > 0; off >>= 1) mx = fmaxf(mx, __shfl_xor(mx, off));
    // lane-cooperative sum of exp (fixed xor-tree -> deterministic)
    float se = 0.0f;
    for (int c = ln; c < CC; c += 32) se += __expf(x[c] - mx);
    for (int off = 16; off > 0; off >>= 1) se += __shfl_xor(se, off);

    if (ln == 0) {
        int p = pos[(size_t)b*AA + a];
        int n = neg[(size_t)b*AA + a];
        int bg = best_idx[(size_t)b*AA + a];
        int m = p ? gt_labels[b*GG + bg] : 0;
        float pt = __expf(x[m] - mx) / se;
        pt = fminf(fmaxf(pt, EPSF), 1.0f - EPSF);
        float at = (m > 0) ? ALPHA_ : (1.0f - ALPHA_);
        float focal = at * (1.0f - pt) * (1.0f - pt) * (-logf(pt));
        w1[wv] = (p || n) ? focal : 0.0f;

        float bloss = 0.0f;
        if (p) {
            float4 an = ((const float4*)anchors)[a];
            float4 gb = ((const float4*)gt_boxes)[(size_t)b*GG + bg];
            const float inv = 1.0f / INPUT_SIZE;
            float m0 = gb.x*inv, m1 = gb.y*inv, m2 = gb.z*inv, m3 = gb.w*inv;
            float gcx = 0.5f*(m0+m2), gcy = 0.5f*(m1+m3);
            float gw = fmaxf(m2-m0, EPSF), gh = fmaxf(m3-m1, EPSF);
            float t0 = (gcx-an.x)/(an.z*VAR_XY);
            float t1 = (gcy-an.y)/(an.w*VAR_XY);
            float t2 = logf(gw/an.z)/VAR_WH;
            float t3 = logf(gh/an.w)/VAR_WH;
            float4 bp = ((const float4*)box_preds)[(size_t)b*AA + a];
            float d[4] = { bp.x-t0, bp.y-t1, bp.z-t2, bp.w-t3 };
            for (int i = 0; i < 4; ++i) {
                float dd = fabsf(d[i]);
                bloss += (dd < 1.0f) ? 0.5f*dd*dd : dd - 0.5f;
            }
        }
        w2[wv] = bloss;
    }
    __syncthreads();
    if (threadIdx.x == 0) {
        float c = 0.0f, bx = 0.0f;
        for (int i = 0; i < 8; ++i) { c += w1[i]; bx += w2[i]; }
        cls_part[(size_t)b*NB_CB + blockIdx.x] = c;
        box_part[(size_t)b*NB_CB + blockIdx.x] = bx;
    }
}

// ================= kernel 5: exact deterministic top-K selection ==============
__global__ void topk_kernel(const float* __restrict__ best_iou,
                            const int*   __restrict__ pos,
                            const int*   __restrict__ best_idx,
                            const float* __restrict__ mask_coeffs,
                            int*   __restrict__ sel_idx,
                            float* __restrict__ sel_w,
                            int*   __restrict__ sel_mi,
                            float* __restrict__ sel_coef) {
    int b = blockIdx.x; int tid = threadIdx.x;
    __shared__ unsigned char selflag[AA];
    __shared__ unsigned long long red[256];
    __shared__ int sidx[KPOS];
    for (int i = tid; i < AA; i += 256) selflag[i] = 0;
    __syncthreads();
    for (int k = 0; k < KPOS; ++k) {
        unsigned long long best = 0;
        for (int a = tid; a < AA; a += 256) {
            if (selflag[a]) continue;
            float sc = pos[(size_t)b*AA + a] ? (best_iou[(size_t)b*AA + a] + 1.0f) : 0.0f;
            unsigned long long key =
                ((unsigned long long)__float_as_uint(sc) << 32) |
                (unsigned long long)(0x7FFFFFFFu - (unsigned)a);
            if (key > best) best = key;
        }
        red[tid] = best; __syncthreads();
        // tree reduce: u64 max is order-independent -> deterministic
        for (int off = 128; off > 0; off >>= 1) {
            if (tid < off) { if (red[tid + off] > red[tid]) red[tid] = red[tid + off]; }
            __syncthreads();
        }
        if (tid == 0) {
            unsigned long long m = red[0];
            int a = (int)(0x7FFFFFFFu - (unsigned)(m & 0xFFFFFFFFull));
            selflag[a] = 1;
            sidx[k] = a;
            sel_idx[b*KPOS + k] = a;
            sel_w[b*KPOS + k] = pos[(size_t)b*AA + a] ? 1.0f : 0.0f;
            sel_mi[b*KPOS + k] = best_idx[(size_t)b*AA + a];
        }
        __syncthreads();
    }
    for (int i = tid; i < KPOS*PP; i += 256) {
        int k = i / PP, p = i % PP;
        sel_coef[(size_t)b*KPOS*PP + i] =
            mask_coeffs[((size_t)b*AA + sidx[k])*PP + p];
    }
}

// ================= kernel 6: mask logits via V_WMMA_F32_16X16X4_F32 + BCE =====
__global__ void mask_wmma_kernel(const float* __restrict__ protos,
                                 const float* __restrict__ gt_masks,
                                 const float* __restrict__ sel_coef,
                                 const float* __restrict__ sel_w,
                                 const int*   __restrict__ sel_mi,
                                 float* __restrict__ rowsum) {
    int rowtile = blockIdx.x;      // 0..7  (16 rows each => K=128)
    int b = blockIdx.y;            // image
    int tid = threadIdx.x;         // 128 threads = 4 waves
    int wave = tid >> 5;
    int lane = tid & 31;
    int half = lane >> 4;          // 0: lanes 0-15, 1: lanes 16-31
    int t = lane & 15;

    __shared__ float coef[16][PP];
    __shared__ int   mi[16];
    __shared__ float wk[16];
    __shared__ float partial[4][32][8];

    if (tid < 16) {
        int kg = rowtile*16 + tid;
        mi[tid] = sel_mi[b*KPOS + kg];
        wk[tid] = sel_w[b*KPOS + kg];
    }
    for (int i = tid; i < 16*PP; i += 128) {
        coef[i/PP][i%PP] =
            sel_coef[(size_t)b*KPOS*PP + (size_t)(rowtile*16 + i/PP)*PP + (i%PP)];
    }
    __syncthreads();

    // preload A fragments: lane t holds row M=t; VGPR0 k=4s+2*half, VGPR1 k+1
    v2f af[8];
#pragma unroll
    for (int s = 0; s < 8; ++s) {
        af[s].x = coef[t][4*s + 2*half + 0];
        af[s].y = coef[t][4*s + 2*half + 1];
    }
    // uniform base pointers + 32-bit element offsets (SADDR + voffset codegen)
    const float* pb  = protos   + (size_t)b*PP*HW;      // uniform per block
    const float* gmb = gt_masks + (size_t)b*GG*HW;      // uniform per block
    int poff0 = (4*0 + 2*half) * HW;                    // per-lane k row offsets
    int moff[8];
#pragma unroll
    for (int r = 0; r < 8; ++r) moff[r] = mi[r + 8*half] * HW;

    float acc[8];
#pragma unroll
    for (int r = 0; r < 8; ++r) acc[r] = 0.0f;

    for (int ct = wave; ct < NCT; ct += 4) {
        int col  = ct*16 + t;
        int colc = (col < HW) ? col : (HW - 1);     // clamp: loads always legal
        float inbf = (col < HW) ? 1.0f : 0.0f;
        // batch all B-fragment + target loads: 32-bit offsets, one clause
        v2f bf[8];
#pragma unroll
        for (int s = 0; s < 8; ++s) {
            int o = poff0 + 4*s*HW + colc;
            bf[s].x = pb[o];
            bf[s].y = pb[o + HW];
        }
        float tg[8];
#pragma unroll
        for (int r = 0; r < 8; ++r) tg[r] = gmb[moff[r] + colc];
        // chained fp32 WMMAs: 16x16 tile, K=32 in 8 steps of 4
        v8f c = {0.f,0.f,0.f,0.f,0.f,0.f,0.f,0.f};
#pragma unroll
        for (int s = 0; s < 8; ++s)
            c = __builtin_amdgcn_wmma_f32_16x16x4_f32(
                    false, af[s], false, bf[s], (short)0, c, false, false);
#pragma unroll
        for (int r = 0; r < 8; ++r) {
            float l = c[r];
            float bce = fmaxf(l, 0.0f) - l*tg[r] + log1pf(__expf(-fabsf(l)));
            acc[r] += inbf * bce;
        }
    }
#pragma unroll
    for (int r = 0; r < 8; ++r) partial[wave][lane][r] = acc[r];
    __syncthreads();

    // deterministic fixed-order row reduction: thread m handles row m
    if (tid < 16) {
        int m = tid;
        int r = m & 7;
        int l0 = (m < 8) ? 0 : 16;
        float s = 0.0f;
        for (int w = 0; w < 4; ++w)
            for (int l = l0; l < l0 + 16; ++l)
                s += partial[w][l][r];
        rowsum[b*KPOS + rowtile*16 + m] = s * wk[m] * (1.0f / (float)HW);
    }
}

// ================= kernel 7: finalize (single thread, fixed order) ============
__global__ void finalize_kernel(const float* __restrict__ cls_part,
                                const float* __restrict__ box_part,
                                const float* __restrict__ rowsum,
                                const int*   __restrict__ numpos,
                                float* __restrict__ out) {
    if (threadIdx.x != 0 || blockIdx.x != 0) return;
    float cls = 0.0f, box = 0.0f, msk = 0.0f;
    for (int i = 0; i < BB*NB_CB; ++i) cls += cls_part[i];
    for (int i = 0; i < BB*NB_CB; ++i) box += box_part[i];
    for (int i = 0; i < BB*KPOS; ++i) msk += rowsum[i];
    int np = 0;
    for (int i = 0; i < BB; ++i) np += numpos[i];
    float denom = (float)((np > 1) ? np : 1);
    out[0] = (CLS_W*cls + BOX_W*box + MASK_W*msk) / denom;
}

// ================= launcher ===================================================
extern "C" void kernel_launch(void* const* d_in, const int* in_sizes, int n_in,
                              void* d_out, int out_size, void* d_ws, size_t ws_size,
                              hipStream_t stream) {
    const float* class_preds = (const float*)d_in[0];
    const float* box_preds   = (const float*)d_in[1];
    const float* mask_coeffs = (const float*)d_in[2];
    const float* prototypes  = (const float*)d_in[3];
    const float* anchors     = (const float*)d_in[4];
    const float* gt_boxes    = (const float*)d_in[5];
    const int*   gt_labels   = (const int*)d_in[6];
    const float* gt_masks    = (const float*)d_in[7];
    float* out = (float*)d_out;

    char* ws = (char*)d_ws;
    unsigned long long* packed = (unsigned long long*)(ws + OFF_PACKED);
    float* best_iou = (float*)(ws + OFF_BIOU);
    int*   best_idx = (int*)  (ws + OFF_BIDX);
    int*   pos      = (int*)  (ws + OFF_POS);
    int*   neg      = (int*)  (ws + OFF_NEG);
    int*   numpos   = (int*)  (ws + OFF_NUMPOS);
    float* cls_part = (float*)(ws + OFF_CLSPART);
    float* box_part = (float*)(ws + OFF_BOXPART);
    int*   sel_idx  = (int*)  (ws + OFF_SELIDX);
    float* sel_w    = (float*)(ws + OFF_SELW);
    int*   sel_mi   = (int*)  (ws + OFF_SELMI);
    float* sel_coef = (float*)(ws + OFF_SELCOEF);
    float* rowsum   = (float*)(ws + OFF_ROWSUM);

    hipMemsetAsync(d_ws, 0, WS_USED, stream);

    match_kernel<<<dim3(NBLK, BB), 256, 0, stream>>>(
        anchors, gt_boxes, best_iou, best_idx, pos, packed);

    scatter_kernel<<<BB, 1, 0, stream>>>(packed, pos, best_idx);

    scan_kernel<<<BB, 256, 0, stream>>>(best_iou, pos, neg, numpos);

    clsbox_kernel<<<dim3(NB_CB, BB), 256, 0, stream>>>(
        class_preds, box_preds, anchors, gt_boxes, gt_labels,
        best_idx, pos, neg, cls_part, box_part);

    topk_kernel<<<BB, 256, 0, stream>>>(
        best_iou, pos, best_idx, mask_coeffs, sel_idx, sel_w, sel_mi, sel_coef);

    mask_wmma_kernel<<<dim3(8, BB), 128, 0, stream>>>(
        prototypes, gt_masks, sel_coef, sel_w, sel_mi, rowsum);

    finalize_kernel<<<1, 1, 0, stream>>>(cls_part, box_part, rowsum, numpos, out);
}